// TriAffine_9302899163642
// MI455X (gfx1250) — compile-verified
//
#include <hip/hip_runtime.h>

// TriAffine on MI455X (gfx1250, wave32, WMMA).
// Path:  W2[b,z,i,j] = z @ W   (GEMM, WMMA f32 16x16x4)
//        T_x[j,z]    = xb . W2 (per-(b,x) matvec, VALU, -> 132KB LDS tile)
//        out[y,z]    = yb @ T_x (WMMA f32 16x16x4 from LDS)
// fp32 throughout to match reference accumulation precision; stage 3 is
// near the 134MB store bandwidth floor so lower precision buys nothing.

typedef __attribute__((ext_vector_type(2))) float v2f;
typedef __attribute__((ext_vector_type(4))) float v4f;
typedef __attribute__((ext_vector_type(8))) float v8f;

#define BD   2
#define LL   256
#define NIN  128
#define NB   129      // n_in + bias
#define JP   132      // padded j (multiple of 4)

// ---------------------------------------------------------------------------
// Stage 1: W2[(b,z), i, j] = sum_k zin[(b,z), k] * W[i, k, j]
// grid = (jtile 9, mtile 32, i 129), block = 1 wave (32 threads).
// Each wave: 16(M=bz) x 16(N=j) tile, K=128 -> 32 x wmma_f32_16x16x4_f32.
// Branch-free OOB handling: clamp column address, multiply by 0/1 mask, so
// the K-loop is pure global_load + wmma (no exec-mask churn).
// ---------------------------------------------------------------------------
__global__ void __launch_bounds__(32)
triaffine_stage1(const float* __restrict__ zin,   // [B*L, 128]
                 const float* __restrict__ wgt,   // [129, 128, 129]
                 float*       __restrict__ w2)    // [B*L, 129, JP]
{
    const int jt   = blockIdx.x;          // 0..8
    const int mt   = blockIdx.y;          // 0..31
    const int i    = blockIdx.z;          // 0..128
    const int lane = threadIdx.x;
    const int half = lane >> 4;           // 0/1
    const int l16  = lane & 15;

    const int  j    = jt * 16 + l16;      // N index (j)
    const int  m    = mt * 16 + l16;      // M index (b*L+z), always < 512
    const int  kb   = half * 2;           // per-lane K sub-offset
    const bool jok  = (j < NB);
    const int  jc   = jok ? j : (NB - 1); // clamped: loads always in-bounds
    const float msk = jok ? 1.0f : 0.0f;

    const float* wbase = wgt + (size_t)i * NIN * NB;
    v8f acc = {};
    for (int ks = 0; ks < NIN; ks += 4) {
        v2f a, bm;
        // A: 16x4 f32 tile of zin  (lane=M row; VGPR0=K kb, VGPR1=K kb+1)
        a.x = zin[(size_t)m * NIN + ks + kb];
        a.y = zin[(size_t)m * NIN + ks + kb + 1];
        // B: 4x16 f32 tile of W[i]  (lane=N col; rows kb, kb+1)
        const float* wp = wbase + (size_t)(ks + kb) * NB + jc;
        bm.x = msk * wp[0];
        bm.y = msk * wp[NB];
        acc = __builtin_amdgcn_wmma_f32_16x16x4_f32(
            false, a, false, bm, (short)0, acc, false, false);
    }
    if (jok) {
        const int rbase = mt * 16 + half * 8;   // C/D: vgpr r -> row rbase+r
        #pragma unroll
        for (int r = 0; r < 8; ++r)
            w2[((size_t)(rbase + r) * NB + i) * JP + j] = acc[r];
    }
}

// ---------------------------------------------------------------------------
// Stages 2+3 fused: one workgroup per (b, x).  256 threads (8 waves).
// LDS (dynamic, 271KB < 320KB/WGP):
//   Tm[132][256] : T_x(j,z) tile, rows 129..131 zero-padded
//   Yl[256][132] : yb(y,j) staged, cols 129..131 zero-padded
//   Xl[132]      : xb(x,:) row
// ---------------------------------------------------------------------------
__global__ void __launch_bounds__(256)
triaffine_stage23(const float* __restrict__ xin,  // [B, L, 128]
                  const float* __restrict__ yin,  // [B, L, 128]
                  const float* __restrict__ w2,   // [B*L(z), 129, JP]
                  float*       __restrict__ out)  // [B, L(x), L(y), L(z)]
{
    extern __shared__ float smem[];
    float* Tm = smem;                         // 132*256
    float* Yl = smem + JP * LL;               // 256*132
    float* Xl = smem + JP * LL + LL * JP;     // 132

    const int bid = blockIdx.x;
    const int b   = bid >> 8;
    const int xi  = bid & 255;
    const int tid = threadIdx.x;

    // ---- stage xb row (with bias, zero pad) into LDS -----------------------
    if (tid < JP) {
        float v = 0.0f;
        if (tid < NIN)       v = xin[((size_t)b * LL + xi) * NIN + tid];
        else if (tid == NIN) v = 1.0f;
        Xl[tid] = v;
    }
    // ---- stage yb tile [256][132] into LDS ---------------------------------
    for (int e = tid; e < LL * JP; e += 256) {
        const int yy = e / JP, j = e % JP;
        float v = 0.0f;
        if (j < NIN)       v = yin[((size_t)b * LL + yy) * NIN + j];
        else if (j == NIN) v = 1.0f;
        Yl[e] = v;
    }
    __syncthreads();

    // ---- stage 2: Tm[j][z] = sum_i Xl[i] * w2[(b*L+z)*129 + i][j] ----------
    // Lane jt covers j = 4*jt..4*jt+3 via one float4 (global_load_b128,
    // coalesced 512B per (z,i) row).  j==128 bias column is wave-uniform.
    const int jt = tid & 31;
    const int zt = tid >> 5;                  // 0..7
    for (int zz = 0; zz < 32; ++zz) {
        const int z = zz * 8 + zt;
        const float* wz = w2 + (size_t)(b * LL + z) * NB * JP;
        // prefetch next z-row block of W2 (L2-resident) toward the WGP
        if (zz < 31)
            __builtin_prefetch(wz + (size_t)8 * NB * JP + 4 * jt, 0, 1);
        v4f  a03 = {};
        float a4 = 0.0f;
        for (int i = 0; i < NB; ++i) {
            const float a  = Xl[i];
            const float* wp = wz + (size_t)i * JP;
            const v4f  w4 = *(const v4f*)(wp + 4 * jt);   // j = 4jt..4jt+3
            a03 += a * w4;
            a4  += a * wp[128];               // uniform; stored by jt==0 only
        }
        #pragma unroll
        for (int c = 0; c < 4; ++c)
            Tm[(4 * jt + c) * LL + z] = a03[c];
        if (jt == 0) Tm[128 * LL + z] = a4;
    }
    // zero-pad K rows 129..131 so the WMMA K-loop can run K=132 cleanly
    #pragma unroll
    for (int j = NB; j < JP; ++j) Tm[j * LL + tid] = 0.0f;
    __syncthreads();

    // ---- stage 3: out[y,z] = yb(256x132) @ Tm(132x256), WMMA from LDS ------
    const int lane = tid & 31;
    const int wv   = tid >> 5;                // 0..7
    const int half = lane >> 4;
    const int l16  = lane & 15;
    float* obase = out + (size_t)(b * LL + xi) * LL * LL;   // [y][z] block

    for (int t = wv; t < 256; t += 8) {       // 16x16 tiles of the 256x256 out
        const int yt   = t >> 4;
        const int ztl  = t & 15;
        const int arow = yt  * 16 + l16;
        const int bcol = ztl * 16 + l16;
        v8f acc = {};
        #pragma unroll 4
        for (int s = 0; s < 33; ++s) {        // K = 132 = 33 * 4
            const int kb = s * 4 + half * 2;
            v2f a, bm;
            a.x  = Yl[arow * JP + kb];
            a.y  = Yl[arow * JP + kb + 1];
            bm.x = Tm[(kb    ) * LL + bcol];
            bm.y = Tm[(kb + 1) * LL + bcol];
            acc = __builtin_amdgcn_wmma_f32_16x16x4_f32(
                false, a, false, bm, (short)0, acc, false, false);
        }
        const int rb = yt * 16 + half * 8;
        #pragma unroll
        for (int r = 0; r < 8; ++r)
            obase[(size_t)(rb + r) * LL + bcol] = acc[r];
    }
}

// ---------------------------------------------------------------------------
extern "C" void kernel_launch(void* const* d_in, const int* in_sizes, int n_in,
                              void* d_out, int out_size, void* d_ws, size_t ws_size,
                              hipStream_t stream) {
    (void)in_sizes; (void)n_in; (void)out_size; (void)ws_size;
    const float* x = (const float*)d_in[0];   // [2,256,128]
    const float* y = (const float*)d_in[1];   // [2,256,128]
    const float* z = (const float*)d_in[2];   // [2,256,128]
    const float* w = (const float*)d_in[3];   // [129,128,129,1]
    float* out = (float*)d_out;               // [2,256,256,256]
    float* w2  = (float*)d_ws;                // 512*129*132*4 = 34.9 MB scratch

    dim3 g1(9, 32, 129);                      // jtile x mtile x i
    triaffine_stage1<<<g1, 32, 0, stream>>>(z, w, w2);

    const size_t lds = (size_t)(JP * LL + LL * JP + JP) * sizeof(float); // 271KB
    triaffine_stage23<<<dim3(BD * LL), 256, lds, stream>>>(x, y, w2, out);
}